// SSIM_53128745451912
// MI455X (gfx1250) — compile-verified
//
#include <hip/hip_runtime.h>

// ---------------- problem constants ----------------
#define B_    32
#define H_    640
#define W_    368
#define OH_   634             // H - 6
#define OW_   362             // W - 6
#define NPIX_ (H_ * W_)       // 235520
#define NOUT_TOTAL 7344256.0f // 32 * 634 * 362

// ---------------- tiling ----------------
#define TW   112   // output columns per workgroup (7 waves x 16)
#define IW   118   // input columns actually needed (TW + 6)
#define LDW  122   // LDS column stride (pad; avoids bank clash on strided row reads)
#define IR   22    // input rows per band (16 + 6)
#define ORR  16    // output rows per band
#define NWG  5120  // 4 * 40 * 32 workgroups in main kernel

typedef float v2f __attribute__((ext_vector_type(2)));
typedef float v8f __attribute__((ext_vector_type(8)));

#ifndef __has_builtin
#define __has_builtin(x) 0
#endif

#if __has_builtin(__builtin_amdgcn_global_load_async_to_lds_b32)
typedef __attribute__((address_space(1))) int as1_int;   // global
typedef __attribute__((address_space(3))) int as3_int;   // LDS

__device__ __forceinline__ void async_ld_b32(const float* g, float* l) {
    __builtin_amdgcn_global_load_async_to_lds_b32(
        (as1_int*)const_cast<float*>(g), (as3_int*)l, 0, 0);
}
#define HAVE_ASYNC_LDS 1
#else
#define HAVE_ASYNC_LDS 0
#endif

// ---------------- kernel 1: per-batch max of target (8 partials per batch) ----------------
__global__ __launch_bounds__(256) void bmax_kernel(const float* __restrict__ tgt,
                                                   float* __restrict__ pmax) {
    __shared__ float red[256];
    const int b = blockIdx.y, chunk = blockIdx.x, tid = threadIdx.x;
    const float4* t4 = (const float4*)(tgt + (size_t)b * NPIX_);
    const int n4  = NPIX_ / 4;   // 58880
    const int per = n4 / 8;      // 7360
    const int beg = chunk * per, end = beg + per;
    float m = 0.f;               // inputs are uniform [0,1): nonnegative
    for (int i = beg + tid; i < end; i += 256) {
        float4 v = t4[i];
        m = fmaxf(m, fmaxf(fmaxf(v.x, v.y), fmaxf(v.z, v.w)));
    }
    red[tid] = m;
    __syncthreads();
    for (int s = 128; s > 0; s >>= 1) {
        if (tid < s) red[tid] = fmaxf(red[tid], red[tid + s]);
        __syncthreads();
    }
    if (tid == 0) pmax[b * 8 + chunk] = red[0];
}

// ---------------- kernel 2: SSIM tiles; horizontal 7-tap via WMMA f32 16x16x4 ----------------
__global__ __launch_bounds__(224) void ssim_kernel(const float* __restrict__ X,
                                                   const float* __restrict__ Y,
                                                   const float* __restrict__ pmax,
                                                   float* __restrict__ partial) {
    __shared__ float rawX[IR * LDW];
    __shared__ float rawY[IR * LDW];
    __shared__ float TS[5][ORR * LDW];   // vertical box sums: Sx,Sy,Sxx,Syy,Sxy
    __shared__ float red[224];

    const int tid     = threadIdx.x;
    const int cb      = blockIdx.x;      // 0..3
    const int band    = blockIdx.y;      // 0..39
    const int b       = blockIdx.z;      // 0..31
    const int colBase = cb * TW;
    const int rowBase = band * ORR;

    // per-batch data range -> C1, C2
    float dr = 0.f;
#pragma unroll
    for (int j = 0; j < 8; ++j) dr = fmaxf(dr, pmax[b * 8 + j]);
    const float C1 = (0.01f * dr) * (0.01f * dr);
    const float C2 = (0.03f * dr) * (0.03f * dr);

    // ---- phase 1a: stage raw X/Y tile into LDS ----
    // Out-of-range elements load a CLAMPED (finite) address instead of a zero
    // fill: such elements only ever feed K-lanes where the band matrix is 0
    // (finite * 0 == 0; clamping avoids the WMMA 0*Inf->NaN rule on garbage)
    // or output pixels masked out of the reduction, so results are unchanged.
    const float* Xb = X + (size_t)b * NPIX_;
    const float* Yb = Y + (size_t)b * NPIX_;
#if HAVE_ASYNC_LDS
    // CDNA5 async global->LDS path: no VGPR round-trip, tracked by ASYNCcnt.
    for (int i = tid; i < IR * LDW; i += 224) {
        const int r = i / LDW, c = i - r * LDW;
        const int gc = min(colBase + c, W_ - 1);
        const int gr = min(rowBase + r, H_ - 1);
        const size_t off = (size_t)gr * W_ + gc;
        async_ld_b32(Xb + off, &rawX[i]);
        async_ld_b32(Yb + off, &rawY[i]);
    }
#if __has_builtin(__builtin_amdgcn_s_wait_asynccnt)
    __builtin_amdgcn_s_wait_asynccnt(0);
#else
    asm volatile("s_wait_asynccnt 0x0" ::: "memory");
#endif
#else
    for (int i = tid; i < IR * LDW; i += 224) {
        const int r = i / LDW, c = i - r * LDW;
        const int gc = min(colBase + c, W_ - 1);
        const int gr = min(rowBase + r, H_ - 1);
        const size_t off = (size_t)gr * W_ + gc;
        rawX[i] = Xb[off];
        rawY[i] = Yb[off];
    }
#endif
    __syncthreads();

    // ---- phase 1b: vertical 7-tap running sums, one column per thread ----
    if (tid < LDW) {
        const int c = tid;
        float Sx = 0.f, Sy = 0.f, Sxx = 0.f, Syy = 0.f, Sxy = 0.f;
#pragma unroll
        for (int i = 0; i < IR; ++i) {
            const float x = rawX[i * LDW + c], y = rawY[i * LDW + c];
            Sx += x; Sy += y; Sxx += x * x; Syy += y * y; Sxy += x * y;
            if (i >= 6) {
                const int o = i - 6;
                TS[0][o * LDW + c] = Sx;  TS[1][o * LDW + c] = Sy;
                TS[2][o * LDW + c] = Sxx; TS[3][o * LDW + c] = Syy;
                TS[4][o * LDW + c] = Sxy;
                const float ox = rawX[o * LDW + c], oy = rawY[o * LDW + c];
                Sx -= ox; Sy -= oy; Sxx -= ox * ox; Syy -= oy * oy; Sxy -= ox * oy;
            }
        }
    }
    __syncthreads();

    // ---- phase 2: horizontal 7-tap as D += A(16x4 data) x B(4x16 ones-band) ----
    const int lane  = tid & 31;
    const int wv    = tid >> 5;          // 0..6 -> one 16x16 tile per wave
    const int n     = lane & 15;         // output column within tile (B/D layout)
    const int kh    = lane >> 4;         // K-half selector (A/B layout)
    const int tbase = wv * 16;
    const int arow  = (lane & 15) * LDW; // A-matrix row M = lane & 15

    v8f acc0 = {}, acc1 = {}, acc2 = {}, acc3 = {}, acc4 = {};
#pragma unroll
    for (int k0 = 0; k0 < 24; k0 += 4) {       // K = 22 padded to 24
        const int kv = k0 + 2 * kh;
        v2f bm;
        bm.x = (kv     >= n && kv     <= n + 6 && kv     < 22) ? 1.f : 0.f;
        bm.y = (kv + 1 >= n && kv + 1 <= n + 6 && kv + 1 < 22) ? 1.f : 0.f;
        const int col = tbase + kv;
        v2f a;
        a.x = TS[0][arow + col]; a.y = TS[0][arow + col + 1];
        acc0 = __builtin_amdgcn_wmma_f32_16x16x4_f32(false, a, false, bm, (short)0, acc0, false, false);
        a.x = TS[1][arow + col]; a.y = TS[1][arow + col + 1];
        acc1 = __builtin_amdgcn_wmma_f32_16x16x4_f32(false, a, false, bm, (short)0, acc1, false, false);
        a.x = TS[2][arow + col]; a.y = TS[2][arow + col + 1];
        acc2 = __builtin_amdgcn_wmma_f32_16x16x4_f32(false, a, false, bm, (short)0, acc2, false, false);
        a.x = TS[3][arow + col]; a.y = TS[3][arow + col + 1];
        acc3 = __builtin_amdgcn_wmma_f32_16x16x4_f32(false, a, false, bm, (short)0, acc3, false, false);
        a.x = TS[4][arow + col]; a.y = TS[4][arow + col + 1];
        acc4 = __builtin_amdgcn_wmma_f32_16x16x4_f32(false, a, false, bm, (short)0, acc4, false, false);
    }

    // ---- pointwise SSIM on D-layout elements (row = i + 8*kh, col = n) ----
    const float inv49 = 1.f / 49.f;
    const float covn  = 49.f / 48.f;
    float lsum = 0.f;
#pragma unroll
    for (int i = 0; i < 8; ++i) {
        const int grow = rowBase + i + 8 * kh;
        const int gcol = colBase + tbase + n;
        const float sx = acc0[i], sy = acc1[i], sxx = acc2[i], syy = acc3[i], sxy = acc4[i];
        const float ux = sx * inv49, uy = sy * inv49;
        const float vx  = covn * (sxx * inv49 - ux * ux);
        const float vy  = covn * (syy * inv49 - uy * uy);
        const float vxy = covn * (sxy * inv49 - ux * uy);
        const float A1 = 2.f * ux * uy + C1;
        const float A2 = 2.f * vxy + C2;
        const float B1 = ux * ux + uy * uy + C1;
        const float B2 = vx + vy + C2;
        const float S = (A1 * A2) / (B1 * B2);
        if (grow < OH_ && gcol < OW_) lsum += S;
    }

    // ---- workgroup reduction -> one partial (deterministic tree) ----
    red[tid] = lsum;
    __syncthreads();
    for (int s = 128; s > 0; s >>= 1) {
        if (tid < s && tid + s < 224) red[tid] += red[tid + s];
        __syncthreads();
    }
    if (tid == 0) {
        const int wg = (b * gridDim.y + band) * gridDim.x + cb;
        partial[wg] = red[0];
    }
}

// ---------------- kernel 3: deterministic final reduction ----------------
__global__ __launch_bounds__(256) void finalize_kernel(const float* __restrict__ partial,
                                                       float* __restrict__ out) {
    __shared__ float red[256];
    const int tid = threadIdx.x;
    float s = 0.f;
    for (int i = tid; i < NWG; i += 256) s += partial[i];
    red[tid] = s;
    __syncthreads();
    for (int st = 128; st > 0; st >>= 1) {
        if (tid < st) red[tid] += red[tid + st];
        __syncthreads();
    }
    if (tid == 0) out[0] = 1.f - red[0] * (1.f / NOUT_TOTAL);
}

// ---------------- host entry ----------------
extern "C" void kernel_launch(void* const* d_in, const int* in_sizes, int n_in,
                              void* d_out, int out_size, void* d_ws, size_t ws_size,
                              hipStream_t stream) {
    (void)in_sizes; (void)n_in; (void)out_size; (void)ws_size;
    const float* Xp = (const float*)d_in[0];   // 'output'
    const float* Yp = (const float*)d_in[1];   // 'target'
    float* wsf     = (float*)d_ws;
    float* pmax    = wsf;          // 256 floats
    float* partial = wsf + 256;    // 5120 floats

    bmax_kernel    <<<dim3(8, 32),     dim3(256), 0, stream>>>(Yp, pmax);
    ssim_kernel    <<<dim3(4, 40, 32), dim3(224), 0, stream>>>(Xp, Yp, pmax, partial);
    finalize_kernel<<<1,               dim3(256), 0, stream>>>(partial, (float*)d_out);
}